// SpconvToDenseBEV_34591666602815
// MI455X (gfx1250) — compile-verified
//
#include <hip/hip_runtime.h>
#include <hip/hip_bf16.h>

typedef float v8f __attribute__((ext_vector_type(8)));
typedef float v2f __attribute__((ext_vector_type(2)));

#define IN_C   64
#define OUT_C  128
#define N_Z    8
#define BEV_H  256
#define BEV_W  256
#define PLANE  (BEV_H * BEV_W)          // 65536 (stride between channels in out)
#define BSTRIDE (OUT_C * PLANE)         // 8388608 (stride between batches in out)
#define A_STRIDE 68                     // LDS row stride (floats), bank-conflict-free

// ws layout (ints): [0..7] bin counts, [8..15] bin offsets, [16..23] cursors,
// [24..31] reserved, [32..] perm array (capacity N + 128)

__global__ void init_kernel(float* __restrict__ out, int out_n,
                            int* __restrict__ ws, int permCap) {
    int total = out_n > (permCap + 32) ? out_n : (permCap + 32);
    for (int i = blockIdx.x * blockDim.x + threadIdx.x; i < total;
         i += gridDim.x * blockDim.x) {
        if (i < out_n) out[i] = 0.0f;
        if (i < 32) ws[i] = 0;
        if (i < permCap) ws[32 + i] = -1;
    }
}

__global__ void hist_kernel(const int* __restrict__ z, int n, int* __restrict__ ws) {
    __shared__ int h[N_Z];
    if (threadIdx.x < N_Z) h[threadIdx.x] = 0;
    __syncthreads();
    for (int i = blockIdx.x * blockDim.x + threadIdx.x; i < n;
         i += gridDim.x * blockDim.x) {
        int zz = z[i];
        zz = zz < 0 ? 0 : (zz > N_Z - 1 ? N_Z - 1 : zz);
        atomicAdd(&h[zz], 1);
    }
    __syncthreads();
    if (threadIdx.x < N_Z) atomicAdd(&ws[threadIdx.x], h[threadIdx.x]);
}

__global__ void offsets_kernel(int* __restrict__ ws) {
    if (blockIdx.x == 0 && threadIdx.x == 0) {
        int run = 0;
        for (int zz = 0; zz < N_Z; zz++) {
            ws[8 + zz] = run;
            run += (ws[zz] + 15) & ~15;   // align each bin start to 16
        }
    }
}

__global__ void scatter_perm_kernel(const int* __restrict__ z, int n,
                                    int* __restrict__ ws) {
    for (int i = blockIdx.x * blockDim.x + threadIdx.x; i < n;
         i += gridDim.x * blockDim.x) {
        int zz = z[i];
        zz = zz < 0 ? 0 : (zz > N_Z - 1 ? N_Z - 1 : zz);
        int pos = ws[8 + zz] + atomicAdd(&ws[16 + zz], 1);
        ws[32 + pos] = i;
    }
}

// One wave (32-thread block) per 16-point tile (z-uniform by construction).
// The wave owns all 128 output channels: 8 independent 16x16 accumulators
// give 8 parallel WMMA dependency chains to hide XDL latency.
__global__ __launch_bounds__(32) void gemm_scatter_kernel(
    const float* __restrict__ feat, const float* __restrict__ kern,
    const float* __restrict__ bias, const int* __restrict__ bidx,
    const int* __restrict__ hidx, const int* __restrict__ widx,
    const int* __restrict__ zidx, const int* __restrict__ perm,
    float* __restrict__ out) {
    __shared__ __align__(16) float sA[16 * A_STRIDE];
    __shared__ int sCell[16];
    __shared__ int sPt[16];
    __shared__ int sZ;

    int lid = threadIdx.x;           // 0..31, one wave
    if (lid == 0) sZ = -1;
    __syncthreads();                 // single-wave WG: barrier -> S_NOP

    if (lid < 16) {
        int p = perm[blockIdx.x * 16 + lid];
        sPt[lid] = p;
        if (p >= 0) {
            int b = bidx[p], h = hidx[p], w = widx[p];
            sCell[lid] = b * BSTRIDE + h * BEV_W + w;
            int zz = zidx[p];
            zz = zz < 0 ? 0 : (zz > N_Z - 1 ? N_Z - 1 : zz);
            sZ = zz;                 // benign race: tile is z-uniform
        } else {
            sCell[lid] = -1;
        }
    }
    __syncthreads();

    int zz = sZ;
    if (zz < 0) return;              // wave-uniform: padding-only tile

    // Gather A tile (16 x 64 f32) into LDS; zero padded rows.
    {
        int row = lid >> 1, half = lid & 1;   // 32 floats per thread
        int p = sPt[row];
        float* dst = &sA[row * A_STRIDE + half * 32];
        if (p >= 0) {
            const float* src = feat + (long long)p * IN_C + half * 32;
#pragma unroll
            for (int q = 0; q < 8; q++)
                *(float4*)(dst + 4 * q) = *(const float4*)(src + 4 * q);
        } else {
            float4 zf = make_float4(0.f, 0.f, 0.f, 0.f);
#pragma unroll
            for (int q = 0; q < 8; q++)
                *(float4*)(dst + 4 * q) = zf;
        }
    }
    __syncthreads();

    const float* kb = kern + (long long)zz * IN_C * OUT_C;
    int halfsel = lid >> 4;          // 0: lanes 0-15, 1: lanes 16-31
    int ln = lid & 15;

    v8f acc[8];
#pragma unroll
    for (int t = 0; t < 8; t++) acc[t] = (v8f){};

    const float* arow = &sA[ln * A_STRIDE + 2 * halfsel];

    for (int ks = 0; ks < IN_C / 4; ks++) {
        // A frag: lane (halfsel,ln) holds M=ln, K = 4*ks + 2*halfsel + {0,1}
        v2f a = *(const v2f*)(arow + 4 * ks);
        const float* kr0 = kb + (4 * ks + 2 * halfsel) * OUT_C + ln;
#pragma unroll
        for (int t = 0; t < 8; t++) {
            v2f b;
            b.x = kr0[t * 16];
            b.y = kr0[OUT_C + t * 16];
            acc[t] = __builtin_amdgcn_wmma_f32_16x16x4_f32(
                false, a, false, b, (short)0, acc[t], false, false);
        }
    }

    // Hoist per-row cell bases and per-channel bias into registers.
    int cells[8];
#pragma unroll
    for (int r = 0; r < 8; r++) cells[r] = sCell[r + 8 * halfsel];
    float bvs[8];
#pragma unroll
    for (int t = 0; t < 8; t++) bvs[t] = bias[t * 16 + ln];

    // C/D layout: VGPR r, lane -> row M = r + 8*halfsel, column N = channel.
#pragma unroll
    for (int r = 0; r < 8; r++) {
        int cb = cells[r];
        if (cb >= 0) {
            float* po = out + cb + ln * PLANE;
#pragma unroll
            for (int t = 0; t < 8; t++) {
                atomicAdd(po + t * 16 * PLANE, acc[t][r] + bvs[t]);
            }
        }
    }
}

extern "C" void kernel_launch(void* const* d_in, const int* in_sizes, int n_in,
                              void* d_out, int out_size, void* d_ws, size_t ws_size,
                              hipStream_t stream) {
    const float* feat = (const float*)d_in[0];
    const float* kern = (const float*)d_in[1];
    const float* bias = (const float*)d_in[2];
    const int* bidx   = (const int*)d_in[3];
    const int* hidx   = (const int*)d_in[4];
    const int* widx   = (const int*)d_in[5];
    const int* zidx   = (const int*)d_in[6];
    float* out = (float*)d_out;
    int* ws = (int*)d_ws;

    const int N = in_sizes[3];                 // 300000 points
    const int permCap = N + 8 * 16;            // bin padding headroom
    const int maxTiles = (N + 15) / 16 + N_Z;  // covers per-bin alignment waste

    init_kernel<<<2048, 256, 0, stream>>>(out, out_size, ws, permCap);
    hist_kernel<<<1024, 256, 0, stream>>>(zidx, N, ws);
    offsets_kernel<<<1, 32, 0, stream>>>(ws);
    scatter_perm_kernel<<<1024, 256, 0, stream>>>(zidx, N, ws);
    gemm_scatter_kernel<<<maxTiles, 32, 0, stream>>>(
        feat, kern, bias, bidx, hidx, widx, zidx, ws + 32, out);
}